// Model_28681791602765
// MI455X (gfx1250) — compile-verified
//
#include <hip/hip_runtime.h>

typedef int   v4i __attribute__((ext_vector_type(4)));
typedef float v4f __attribute__((ext_vector_type(4)));

#define BLOCK  256
#define IPT    16
#define TILE   (BLOCK * IPT)   // 4096 elements per block
#define NWAVES (BLOCK / 32)    // 8 wave32 waves per block
#define ROUNDS (IPT / 4)       // 4 rounds of int4 per thread

// ---------------------------------------------------------------------------
// Pass 1: per-block accepted count via wave32 ballot/popcount.
// Single per-thread address; rounds addressed via immediate offsets.
// ---------------------------------------------------------------------------
__global__ __launch_bounds__(BLOCK) void count_kernel(const int* __restrict__ idx,
                                                      int* __restrict__ blockCounts,
                                                      int n) {
    __shared__ int waveCnt[NWAVES];
    const int tid  = threadIdx.x;
    const int lane = tid & 31;
    const int warp = tid >> 5;
    const int base = (int)blockIdx.x * TILE;

    int cnt = 0;
    if (base + TILE <= n) {
        const v4i* p = (const v4i*)(idx + base + tid * 4);  // one 64-bit address
        #pragma unroll
        for (int r = 0; r < ROUNDS; ++r) {
            v4i t = p[r * BLOCK];                           // IOFFSET = r*4096 bytes
            cnt += __builtin_popcount(__builtin_amdgcn_ballot_w32(t.x >= 0));
            cnt += __builtin_popcount(__builtin_amdgcn_ballot_w32(t.y >= 0));
            cnt += __builtin_popcount(__builtin_amdgcn_ballot_w32(t.z >= 0));
            cnt += __builtin_popcount(__builtin_amdgcn_ballot_w32(t.w >= 0));
        }
    } else {
        #pragma unroll
        for (int r = 0; r < ROUNDS; ++r) {
            const int e = base + r * (BLOCK * 4) + tid * 4;
            v4i t;
            t.x = (e + 0 < n) ? idx[e + 0] : -1;
            t.y = (e + 1 < n) ? idx[e + 1] : -1;
            t.z = (e + 2 < n) ? idx[e + 2] : -1;
            t.w = (e + 3 < n) ? idx[e + 3] : -1;
            cnt += __builtin_popcount(__builtin_amdgcn_ballot_w32(t.x >= 0));
            cnt += __builtin_popcount(__builtin_amdgcn_ballot_w32(t.y >= 0));
            cnt += __builtin_popcount(__builtin_amdgcn_ballot_w32(t.z >= 0));
            cnt += __builtin_popcount(__builtin_amdgcn_ballot_w32(t.w >= 0));
        }
    }
    if (lane == 0) waveCnt[warp] = cnt;   // identical across the wave
    __syncthreads();
    if (tid == 0) {
        int s = 0;
        #pragma unroll
        for (int w = 0; w < NWAVES; ++w) s += waveCnt[w];
        blockCounts[blockIdx.x] = s;
    }
}

// ---------------------------------------------------------------------------
// Pass 2: exclusive scan of block counts, single block.
// Also writes the grand total to counts[numBlocks] for the tail-zero pass.
// ---------------------------------------------------------------------------
__global__ __launch_bounds__(BLOCK) void scan_kernel(int* __restrict__ counts, int numBlocks) {
    __shared__ int waveSum[NWAVES];
    const int tid  = threadIdx.x;
    const int lane = tid & 31;
    const int warp = tid >> 5;

    const int S     = (numBlocks + BLOCK - 1) / BLOCK;
    const int begin = tid * S;
    const int end   = (begin + S < numBlocks) ? (begin + S) : numBlocks;

    int sum = 0;
    for (int i = begin; i < end; ++i) sum += counts[i];

    // inclusive wave32 scan of per-thread sums
    int x = sum;
    #pragma unroll
    for (int off = 1; off < 32; off <<= 1) {
        int y = __shfl_up(x, off, 32);
        if (lane >= off) x += y;
    }
    if (lane == 31) waveSum[warp] = x;
    __syncthreads();

    int waveOff = 0;
    for (int w = 0; w < warp; ++w) waveOff += waveSum[w];

    int run = waveOff + (x - sum);        // exclusive prefix for this thread's range
    for (int i = begin; i < end; ++i) {
        int c = counts[i];
        counts[i] = run;
        run += c;
    }
    // last thread ends with run == grand total (empty ranges also end at total)
    if (tid == BLOCK - 1) counts[numBlocks] = run;
}

// ---------------------------------------------------------------------------
// Zero only the tail [total, n): ~0.3*N floats instead of all N.
// total is a uniform scalar load; below-total threads exit immediately.
// ---------------------------------------------------------------------------
__global__ __launch_bounds__(BLOCK) void tail_zero_kernel(float* __restrict__ out,
                                                          const int* __restrict__ totalPtr,
                                                          int n) {
    const int total = *totalPtr;          // s_load, L2-cached
    const int i = 4 * ((int)blockIdx.x * BLOCK + (int)threadIdx.x);
    if (i + 3 < n) {
        if (i >= total) {
            v4f z = {0.f, 0.f, 0.f, 0.f};
            __builtin_nontemporal_store(z, (v4f*)(out + i));
        } else if (i + 4 > total) {
            for (int j = total; j < i + 4; ++j) out[j] = 0.f;   // boundary straddle
        }
    } else {
        int j = (i > total) ? i : total;
        for (; j < n; ++j) out[j] = 0.f;
    }
}

// ---------------------------------------------------------------------------
// Pass 3: gather + stable scatter. Gathers issued early (the 64MB table is
// L2-resident) so their latency overlaps the LDS prefix computation.
// ---------------------------------------------------------------------------
__global__ __launch_bounds__(BLOCK) void scatter_kernel(const int* __restrict__ idx,
                                                        const float* __restrict__ src,
                                                        float* __restrict__ out,
                                                        const int* __restrict__ blockOffsets,
                                                        int n) {
    __shared__ int waveCnt[ROUNDS][NWAVES];
    const int tid  = threadIdx.x;
    const int lane = tid & 31;
    const int warp = tid >> 5;
    const unsigned ltmask = (1u << lane) - 1u;
    const int base = (int)blockIdx.x * TILE;

    v4i v[ROUNDS];
    v4f g[ROUNDS];
    unsigned bal[ROUNDS][4];

    if (base + TILE <= n) {
        const v4i* p = (const v4i*)(idx + base + tid * 4);  // one 64-bit address
        #pragma unroll
        for (int r = 0; r < ROUNDS; ++r) {
            v4i t = p[r * BLOCK];         // L2 hit (second read of idx)
            v[r] = t;
            g[r].x = src[t.x >= 0 ? t.x : 0];
            g[r].y = src[t.y >= 0 ? t.y : 0];
            g[r].z = src[t.z >= 0 ? t.z : 0];
            g[r].w = src[t.w >= 0 ? t.w : 0];
            bal[r][0] = __builtin_amdgcn_ballot_w32(t.x >= 0);
            bal[r][1] = __builtin_amdgcn_ballot_w32(t.y >= 0);
            bal[r][2] = __builtin_amdgcn_ballot_w32(t.z >= 0);
            bal[r][3] = __builtin_amdgcn_ballot_w32(t.w >= 0);
            if (lane == 0) {
                waveCnt[r][warp] =
                    __builtin_popcount(bal[r][0]) + __builtin_popcount(bal[r][1]) +
                    __builtin_popcount(bal[r][2]) + __builtin_popcount(bal[r][3]);
            }
        }
    } else {
        #pragma unroll
        for (int r = 0; r < ROUNDS; ++r) {
            const int e = base + r * (BLOCK * 4) + tid * 4;
            v4i t;
            t.x = (e + 0 < n) ? idx[e + 0] : -1;
            t.y = (e + 1 < n) ? idx[e + 1] : -1;
            t.z = (e + 2 < n) ? idx[e + 2] : -1;
            t.w = (e + 3 < n) ? idx[e + 3] : -1;
            v[r] = t;
            g[r].x = src[t.x >= 0 ? t.x : 0];
            g[r].y = src[t.y >= 0 ? t.y : 0];
            g[r].z = src[t.z >= 0 ? t.z : 0];
            g[r].w = src[t.w >= 0 ? t.w : 0];
            bal[r][0] = __builtin_amdgcn_ballot_w32(t.x >= 0);
            bal[r][1] = __builtin_amdgcn_ballot_w32(t.y >= 0);
            bal[r][2] = __builtin_amdgcn_ballot_w32(t.z >= 0);
            bal[r][3] = __builtin_amdgcn_ballot_w32(t.w >= 0);
            if (lane == 0) {
                waveCnt[r][warp] =
                    __builtin_popcount(bal[r][0]) + __builtin_popcount(bal[r][1]) +
                    __builtin_popcount(bal[r][2]) + __builtin_popcount(bal[r][3]);
            }
        }
    }
    __syncthreads();

    int running = blockOffsets[blockIdx.x];   // uniform -> scalar load
    #pragma unroll
    for (int r = 0; r < ROUNDS; ++r) {
        int wavePref = 0, roundTot = 0;
        #pragma unroll
        for (int w = 0; w < NWAVES; ++w) {
            int c = waveCnt[r][w];            // broadcast LDS reads
            roundTot += c;
            if (w < warp) wavePref += c;
        }
        const int dstBase = running + wavePref;
        running += roundTot;

        int lanePref = __builtin_popcount(bal[r][0] & ltmask) +
                       __builtin_popcount(bal[r][1] & ltmask) +
                       __builtin_popcount(bal[r][2] & ltmask) +
                       __builtin_popcount(bal[r][3] & ltmask);

        int d = dstBase + lanePref;
        if (v[r].x >= 0) { __builtin_nontemporal_store(g[r].x, out + d); ++d; }
        if (v[r].y >= 0) { __builtin_nontemporal_store(g[r].y, out + d); ++d; }
        if (v[r].z >= 0) { __builtin_nontemporal_store(g[r].z, out + d); ++d; }
        if (v[r].w >= 0) { __builtin_nontemporal_store(g[r].w, out + d); }
    }
}

// ---------------------------------------------------------------------------
extern "C" void kernel_launch(void* const* d_in, const int* in_sizes, int n_in,
                              void* d_out, int out_size, void* d_ws, size_t ws_size,
                              hipStream_t stream) {
    (void)n_in; (void)out_size; (void)ws_size;
    const int*   accept_index = (const int*)d_in[0];
    const float* cache        = (const float*)d_in[1];
    float*       out          = (float*)d_out;
    const int n = in_sizes[0];

    const int numBlocks = (n + TILE - 1) / TILE;   // 4096 for N = 16M
    int* blk = (int*)d_ws;                         // numBlocks+1 ints of scratch

    count_kernel<<<numBlocks, BLOCK, 0, stream>>>(accept_index, blk, n);

    scan_kernel<<<1, BLOCK, 0, stream>>>(blk, numBlocks);

    const int n4    = (n + 3) / 4;
    const int zgrid = (n4 + BLOCK - 1) / BLOCK;
    tail_zero_kernel<<<zgrid, BLOCK, 0, stream>>>(out, blk + numBlocks, n);

    scatter_kernel<<<numBlocks, BLOCK, 0, stream>>>(accept_index, cache, out, blk, n);
}